// GCNLayer_26834955666031
// MI455X (gfx1250) — compile-verified
//
#include <hip/hip_runtime.h>

#define N_NODES 100000
#define N_EDGES 3200000
#define DIM 128
#define TILES (N_NODES / 16)      // 6250, exact
#define WPB 8                     // waves per block in GEMM

typedef float v2f __attribute__((ext_vector_type(2)));
typedef float v8f __attribute__((ext_vector_type(8)));

// ---- zero agg + degs workspace ----
__global__ void zero_ws_kernel(float4* __restrict__ p, int n4) {
    int i = blockIdx.x * blockDim.x + threadIdx.x;
    if (i < n4) p[i] = make_float4(0.f, 0.f, 0.f, 0.f);
}

// ---- in-degree via f32 atomics ----
__global__ void degree_kernel(const int* __restrict__ dst, float* __restrict__ degs) {
    int e = blockIdx.x * blockDim.x + threadIdx.x;
    if (e < N_EDGES) atomicAdd(&degs[dst[e]], 1.0f);
}

// ---- norm = rsqrt(max(deg,1)) ----
__global__ void norm_kernel(const float* __restrict__ degs, float* __restrict__ normv) {
    int i = blockIdx.x * blockDim.x + threadIdx.x;
    if (i < N_NODES) normv[i] = rsqrtf(fmaxf(degs[i], 1.0f));
}

// ---- edge aggregation: one wave32 per edge, lane covers 4 features ----
// agg[dst] += features[src] * norm[src]   (pre-norm folded into the gather)
__global__ void edge_agg_kernel(const float* __restrict__ feat,
                                const int* __restrict__ src,
                                const int* __restrict__ dst,
                                const float* __restrict__ normv,
                                float* __restrict__ agg) {
    int t = blockIdx.x * blockDim.x + threadIdx.x;
    int e = t >> 5;
    int lane = t & 31;
    if (e >= N_EDGES) return;
    int s = src[e];
    int d = dst[e];
    float ns = normv[s];
    float4 f = ((const float4*)(feat + (size_t)s * DIM))[lane];   // 512B/edge coalesced
    float* arow = agg + (size_t)d * DIM + lane * 4;
    atomicAdd(arow + 0, f.x * ns);
    atomicAdd(arow + 1, f.y * ns);
    atomicAdd(arow + 2, f.z * ns);
    atomicAdd(arow + 3, f.w * ns);
}

// ---- out = relu((agg*norm) @ W.T) + features, via V_WMMA_F32_16X16X4_F32 ----
// One wave per 16-node tile. W is staged in LDS (64 KB) with the K dimension
// pair-interleaved: Wt[(k>>1)*256 + n*2 + (k&1)] = W[n][k], so a lane's B
// fragment {B[k][n], B[k+1][n]} is one aligned ds_load_b64 into an even VGPR
// pair — no repacking movs. A tile (16x128, post-norm applied) lives in regs.
__global__ void __launch_bounds__(256) gcn_gemm_kernel(
        const float* __restrict__ agg,
        const float* __restrict__ W,
        const float* __restrict__ feat,
        const float* __restrict__ normv,
        float* __restrict__ out) {
    __shared__ float Wt[DIM * DIM];   // 64 KB, pair-interleaved layout

    // stage: LDS writes fully coalesced; global reads scattered but L2-hot
    #pragma unroll
    for (int i = 0; i < (DIM * DIM) / 256; ++i) {
        int idx = threadIdx.x + i * 256;            // flat LDS dest index
        int k = ((idx >> 8) << 1) | (idx & 1);      // kp = idx>>8, klo = idx&1
        int n = (idx >> 1) & 127;
        Wt[idx] = W[n * DIM + k];                   // Wt holds W transposed
    }
    __syncthreads();

    int wave = threadIdx.x >> 5;
    int lane = threadIdx.x & 31;
    int tile = blockIdx.x * WPB + wave;
    if (tile >= TILES) return;        // wave-uniform: EXEC stays all-ones below

    int row   = lane & 15;            // M index of this lane's A elements
    int khalf = lane >> 4;            // 0: K={0,1}, 1: K={2,3} within each step
    int col   = lane & 15;            // N index within an N-tile
    int node  = tile * 16 + row;
    float nr  = normv[node];

    // preload A slice: afrag[kt] = {A[row][4kt+2h], A[row][4kt+2h+1]} * norm
    v2f afrag[32];
    const float2* arow = (const float2*)(agg + (size_t)node * DIM);
    #pragma unroll
    for (int kt = 0; kt < 32; ++kt) {
        float2 a2 = arow[kt * 2 + khalf];
        afrag[kt].x = a2.x * nr;
        afrag[kt].y = a2.y * nr;
    }

    v8f acc[8] = {};   // 8 N-tiles x 16x16 f32 accumulators (64 VGPRs)

    const v2f* Wp = (const v2f*)Wt;   // float2 view: index = kp*128 + n
    #pragma unroll
    for (int kt = 0; kt < 32; ++kt) {
        int kp = kt * 2 + khalf;      // pair index of {k, k+1} for this lane
        v2f a = afrag[kt];
        #pragma unroll
        for (int nt = 0; nt < 8; ++nt) {
            v2f b = Wp[kp * 128 + nt * 16 + col];   // single ds_load_b64
            acc[nt] = __builtin_amdgcn_wmma_f32_16x16x4_f32(
                false, a, false, b, (short)0, acc[nt], false, false);
        }
    }

    // epilogue: relu + residual; D layout: VGPR r -> M = r + 8*khalf, N = col
    #pragma unroll
    for (int nt = 0; nt < 8; ++nt) {
        #pragma unroll
        for (int r = 0; r < 8; ++r) {
            int m = r + 8 * khalf;
            size_t o = (size_t)(tile * 16 + m) * DIM + nt * 16 + col;
            out[o] = fmaxf(acc[nt][r], 0.0f) + feat[o];
        }
    }
}

extern "C" void kernel_launch(void* const* d_in, const int* in_sizes, int n_in,
                              void* d_out, int out_size, void* d_ws, size_t ws_size,
                              hipStream_t stream) {
    const float* feat  = (const float*)d_in[0];   // [N,128]
    const float* W     = (const float*)d_in[1];   // [128,128]
    const int*   esrc  = (const int*)d_in[2];     // [E]
    const int*   edst  = (const int*)d_in[3];     // [E]
    float*       out   = (float*)d_out;           // [N,128]

    // workspace: [agg N*128][degs N][norm N]  (~52 MB)
    float* agg   = (float*)d_ws;
    float* degs  = agg + (size_t)N_NODES * DIM;
    float* normv = degs + N_NODES;

    int n4 = (N_NODES * (DIM + 1)) / 4;           // zero agg + degs together
    zero_ws_kernel<<<(n4 + 255) / 256, 256, 0, stream>>>((float4*)d_ws, n4);
    degree_kernel<<<(N_EDGES + 255) / 256, 256, 0, stream>>>(edst, degs);
    norm_kernel<<<(N_NODES + 255) / 256, 256, 0, stream>>>(degs, normv);
    edge_agg_kernel<<<(N_EDGES * 32 + 255) / 256, 256, 0, stream>>>(
        feat, esrc, edst, normv, agg);
    gcn_gemm_kernel<<<(TILES + WPB - 1) / WPB, 256, 0, stream>>>(
        agg, W, feat, normv, out);
}